// QCNNModel_65481071397508
// MI455X (gfx1250) — compile-verified
//
#include <hip/hip_runtime.h>
#include <math.h>

typedef float v2f __attribute__((ext_vector_type(2)));
typedef float v8f __attribute__((ext_vector_type(8)));

#define DIM   256
#define NQ    8
#define NL    3
#define PSI_S 260   // psi row stride (floats): 260 % 64 == 4 -> conflict-free b64 reads
#define BT_S  260   // B-tile row stride (floats)
#define S2    257   // kernel2 LDS stride (odd -> conflict-free)

// ============================================================================
// Kernel 1: build the batch-independent circuit unitary U (256x256 complex).
// One wave32 per column k. Amplitude index i = lane*8 + r  (i[7:3]=lane,
// i[2:0]=reg). Wire w owns bit (7-w) of i: wires 0..4 -> lane bits 4..0,
// wires 5..7 -> reg bits 2..0.  RZ is diagonal; RY on wires 0..4 uses
// shfl_xor pairing; CNOTs are lane/register permutations.
// ============================================================================

__device__ __forceinline__ void apply_rz(float re[8], float im[8], int lane,
                                         int w, float th) {
  float ch = cosf(0.5f * th), sh = sinf(0.5f * th);
  // bit==0: *= (ch - i sh) ; bit==1: *= (ch + i sh)
  if (w <= 4) {
    float pi_ = ((lane >> (4 - w)) & 1) ? sh : -sh;
#pragma unroll
    for (int r = 0; r < 8; ++r) {
      float nr = re[r] * ch - im[r] * pi_;
      float ni = re[r] * pi_ + im[r] * ch;
      re[r] = nr; im[r] = ni;
    }
  } else {
    int pb = 7 - w;  // 2,1,0
#pragma unroll
    for (int r = 0; r < 8; ++r) {
      float pi_ = ((r >> pb) & 1) ? sh : -sh;
      float nr = re[r] * ch - im[r] * pi_;
      float ni = re[r] * pi_ + im[r] * ch;
      re[r] = nr; im[r] = ni;
    }
  }
}

__device__ __forceinline__ void apply_ry(float re[8], float im[8], int lane,
                                         int w, float th) {
  float ch = cosf(0.5f * th), sh = sinf(0.5f * th);
  // new0 = ch*a0 - sh*a1 ; new1 = sh*a0 + ch*a1
  if (w <= 4) {
    int m = 1 << (4 - w);
    float f = ((lane >> (4 - w)) & 1) ? sh : -sh;  // bit0: -sh*other ; bit1: +sh*other
#pragma unroll
    for (int r = 0; r < 8; ++r) {
      float orr = __shfl_xor(re[r], m, 32);
      float oii = __shfl_xor(im[r], m, 32);
      re[r] = ch * re[r] + f * orr;
      im[r] = ch * im[r] + f * oii;
    }
  } else {
    int pb = 7 - w;
    int stride = 1 << pb;
#pragma unroll
    for (int r0 = 0; r0 < 8; ++r0) {
      if ((r0 >> pb) & 1) continue;   // folds at compile time (w unrolled in caller)
      int r1 = r0 + stride;
      float a0r = re[r0], a0i = im[r0], a1r = re[r1], a1i = im[r1];
      re[r0] = ch * a0r - sh * a1r;  im[r0] = ch * a0i - sh * a1i;
      re[r1] = sh * a0r + ch * a1r;  im[r1] = sh * a0i + ch * a1i;
    }
  }
}

__device__ __forceinline__ void apply_cnots(float re[8], float im[8], int lane) {
  // c=0: ctrl lane bit4, tgt lane bit3
#pragma unroll
  for (int r = 0; r < 8; ++r) {
    float orr = __shfl_xor(re[r], 8, 32), oii = __shfl_xor(im[r], 8, 32);
    bool sel = (lane & 16) != 0;
    re[r] = sel ? orr : re[r];  im[r] = sel ? oii : im[r];
  }
  // c=2: ctrl lane bit2, tgt lane bit1
#pragma unroll
  for (int r = 0; r < 8; ++r) {
    float orr = __shfl_xor(re[r], 2, 32), oii = __shfl_xor(im[r], 2, 32);
    bool sel = (lane & 4) != 0;
    re[r] = sel ? orr : re[r];  im[r] = sel ? oii : im[r];
  }
  // c=4: ctrl lane bit0, tgt reg bit2 -> swap r <-> r+4 when lane is odd
  {
    bool sel = (lane & 1) != 0;
#pragma unroll
    for (int r = 0; r < 4; ++r) {
      float t0 = re[r], t1 = re[r + 4];
      re[r] = sel ? t1 : t0;  re[r + 4] = sel ? t0 : t1;
      float u0 = im[r], u1 = im[r + 4];
      im[r] = sel ? u1 : u0;  im[r + 4] = sel ? u0 : u1;
    }
  }
  // c=6: ctrl reg bit1, tgt reg bit0 -> swap (2,3) and (6,7) unconditionally
  {
    float t;
    t = re[2]; re[2] = re[3]; re[3] = t;  t = im[2]; im[2] = im[3]; im[3] = t;
    t = re[6]; re[6] = re[7]; re[7] = t;  t = im[6]; im[6] = im[7]; im[7] = t;
  }
}

__global__ __launch_bounds__(256) void k_build_unitary(
    const float* __restrict__ params, float* __restrict__ Ur,
    float* __restrict__ Ui) {
  const int lane = threadIdx.x & 31;
  const int col = blockIdx.x * 8 + (threadIdx.x >> 5);  // 32 blocks * 8 waves = 256 cols

  float re[8], im[8];
#pragma unroll
  for (int r = 0; r < 8; ++r) {
    re[r] = (lane * 8 + r == col) ? 1.f : 0.f;
    im[r] = 0.f;
  }

  for (int l = 0; l < NL; ++l) {
#pragma unroll
    for (int w = 0; w < NQ; ++w) {
      const float* p = params + (l * NQ + w) * 3;
      apply_rz(re, im, lane, w, p[0]);
      apply_ry(re, im, lane, w, p[1]);
      apply_rz(re, im, lane, w, p[2]);
    }
    apply_cnots(re, im, lane);
  }

  // Store column-major: U[col][row], row = lane*8 + r
  float4* ur = (float4*)(Ur + (size_t)col * DIM + lane * 8);
  ur[0] = make_float4(re[0], re[1], re[2], re[3]);
  ur[1] = make_float4(re[4], re[5], re[6], re[7]);
  float4* ui = (float4*)(Ui + (size_t)col * DIM + lane * 8);
  ui[0] = make_float4(im[0], im[1], im[2], im[3]);
  ui[1] = make_float4(im[4], im[5], im[6], im[7]);
}

// ============================================================================
// Kernel 2: A[j][k] = sum_i s_i (Ur[j][i]*Ur[k][i] + Ui[j][i]*Ui[k][i]),
// s_i = +1 for i<128 (wire-0 bit = 0), -1 otherwise. A is real symmetric.
// 16x16 output tile per block; U rows staged in LDS (stride 257).
// ============================================================================
__global__ __launch_bounds__(256) void k_build_A(const float* __restrict__ Ur,
                                                 const float* __restrict__ Ui,
                                                 float* __restrict__ A) {
  extern __shared__ float sm[];
  float* Jr = sm;
  float* Ji = sm + 16 * S2;
  float* Kr = sm + 32 * S2;
  float* Ki = sm + 48 * S2;
  const int jt = (blockIdx.x >> 4) * 16;
  const int kt = (blockIdx.x & 15) * 16;

  for (int idx = threadIdx.x; idx < 4096; idx += 256) {
    int rr = idx >> 8, cc = idx & 255;
    Jr[rr * S2 + cc] = Ur[(size_t)(jt + rr) * DIM + cc];
    Ji[rr * S2 + cc] = Ui[(size_t)(jt + rr) * DIM + cc];
    Kr[rr * S2 + cc] = Ur[(size_t)(kt + rr) * DIM + cc];
    Ki[rr * S2 + cc] = Ui[(size_t)(kt + rr) * DIM + cc];
  }
  __syncthreads();

  const int tj = threadIdx.x >> 4, tk = threadIdx.x & 15;
  float acc = 0.f;
  for (int i = 0; i < 128; ++i)
    acc += Jr[tj * S2 + i] * Kr[tk * S2 + i] + Ji[tj * S2 + i] * Ki[tk * S2 + i];
  for (int i = 128; i < 256; ++i)
    acc -= Jr[tj * S2 + i] * Kr[tk * S2 + i] + Ji[tj * S2 + i] * Ki[tk * S2 + i];
  A[(size_t)(jt + tj) * DIM + kt + tk] = acc;
}

// ============================================================================
// Kernel 3 (hot): per block 128 batch rows. Build psi0 tile in LDS, then
// Y = psi0 @ A via V_WMMA_F32_16X16X4_F32 and z = rowdot(Y, psi0), sigmoid.
// Wave m owns rows 16m..16m+15 and loops all 16 N-chunks; its 64 A-operand
// v2f's live in registers (reused across chunks). B-operand tiles of A are
// staged to LDS using A's symmetry (coalesced row loads).
// WMMA layouts (ISA 7.12.2): A-op: lane -> M=lane&15, K = reg + 2*(lane>=16).
// B-op (assumed mirror): N = lane&15, K = reg + 2*(lane>=16).
// C/D: lane,reg -> M = reg + 8*(lane>=16), N = lane&15.
// ============================================================================
__global__ __launch_bounds__(256) void k_qcnn_main(
    const float* __restrict__ x, const float* __restrict__ Amat,
    const float* __restrict__ weight, float* __restrict__ out) {
  extern __shared__ float sm[];
  float* psi = sm;                 // [128][PSI_S]
  float* Bt  = sm + 128 * PSI_S;   // [16][BT_S]

  const int t = threadIdx.x;
  const int lane = t & 31;
  const int wv = t >> 5;           // wave 0..7
  const int blockBase = blockIdx.x * 128;

  // ---- phase 1: psi0[j] = prod_w (bit_w(j) ? sin(x_w/2) : cos(x_w/2)) ----
  {
    int row = t >> 1;
    int h0 = t & 1;                // wire-0 bit of the columns this thread fills
    const float* xr = x + (size_t)(blockBase + row) * NQ;
    float c[8], s[8];
#pragma unroll
    for (int w = 0; w < 8; ++w) {
      float xv = xr[w];
      c[w] = cosf(0.5f * xv);
      s[w] = sinf(0.5f * xv);
    }
    float p0 = h0 ? s[0] : c[0];
    float* prow = psi + row * PSI_S + h0 * 128;
    for (int j = 0; j < 128; ++j) {
      float p = p0;
#pragma unroll
      for (int w = 1; w < 8; ++w) p *= ((j >> (7 - w)) & 1) ? s[w] : c[w];
      prow[j] = p;
    }
  }
  __syncthreads();

  // ---- phase 2: hoist this wave's 64 WMMA A-operands into registers ----
  const int khalf = (lane >> 4) << 1;          // 0 or 2
  const int lrow = wv * 16 + (lane & 15);      // M row owned by this lane
  float2 aop[64];
#pragma unroll
  for (int ks = 0; ks < 64; ++ks)
    aop[ks] = *(const float2*)(psi + lrow * PSI_S + ks * 4 + khalf);

  float zacc[8];
#pragma unroll
  for (int r = 0; r < 8; ++r) zacc[r] = 0.f;

  // ---- phase 3: 16 N-chunks of 16 columns each ----
  for (int chunk = 0; chunk < 16; ++chunk) {
    const int n0 = chunk * 16;
    {  // stage Bt[n][k] = A[n0+n][k]  (valid since A is symmetric)
      int n = t >> 4;
      int kq = (t & 15) * 16;
      const float4* src = (const float4*)(Amat + (size_t)(n0 + n) * DIM + kq);
      float4* dst = (float4*)(Bt + n * BT_S + kq);
#pragma unroll
      for (int q = 0; q < 4; ++q) dst[q] = src[q];
      if (chunk + 1 < 16) __builtin_prefetch(Amat + (size_t)(n0 + 16 + n) * DIM + kq);
    }
    __syncthreads();

    v8f C = {0.f, 0.f, 0.f, 0.f, 0.f, 0.f, 0.f, 0.f};
#pragma unroll
    for (int ks = 0; ks < 64; ++ks) {
      float2 bv = *(const float2*)(Bt + (lane & 15) * BT_S + ks * 4 + khalf);
      v2f a; a.x = aop[ks].x; a.y = aop[ks].y;
      v2f b; b.x = bv.x;      b.y = bv.y;
      C = __builtin_amdgcn_wmma_f32_16x16x4_f32(false, a, false, b, (short)0, C,
                                                false, false);
    }

    // z partial: C[lane,r] = Y[M][n0 + (lane&15)], M = wv*16 + r + 8*(lane>=16)
#pragma unroll
    for (int r = 0; r < 8; ++r) {
      int rowM = wv * 16 + r + ((lane >> 4) << 3);
      zacc[r] += C[r] * psi[rowM * PSI_S + n0 + (lane & 15)];
    }
    __syncthreads();
  }

  // ---- reduce the 16 N-lanes within each half-wave, then sigmoid ----
#pragma unroll
  for (int r = 0; r < 8; ++r) {
    zacc[r] += __shfl_xor(zacc[r], 1, 32);
    zacc[r] += __shfl_xor(zacc[r], 2, 32);
    zacc[r] += __shfl_xor(zacc[r], 4, 32);
    zacc[r] += __shfl_xor(zacc[r], 8, 32);
  }
  if ((lane & 15) == 0) {
    float wgt = weight[0];
#pragma unroll
    for (int r = 0; r < 8; ++r) {
      int row = blockBase + wv * 16 + r + ((lane >> 4) << 3);
      out[row] = 1.f / (1.f + expf(-wgt * zacc[r]));
    }
  }
}

// ============================================================================
extern "C" void kernel_launch(void* const* d_in, const int* in_sizes, int n_in,
                              void* d_out, int out_size, void* d_ws,
                              size_t ws_size, hipStream_t stream) {
  const float* x = (const float*)d_in[0];       // [65536][8]
  const float* params = (const float*)d_in[1];  // [3][8][3]
  const float* weight = (const float*)d_in[2];  // scalar
  float* out = (float*)d_out;                   // [65536]

  // Workspace layout (needs 3 * 256*256 * 4 = 768 KB):
  float* Ur = (float*)d_ws;
  float* Ui = Ur + DIM * DIM;
  float* A  = Ui + DIM * DIM;

  k_build_unitary<<<32, 256, 0, stream>>>(params, Ur, Ui);

  size_t sm2 = (size_t)64 * S2 * sizeof(float);  // 65,792 B
  hipFuncSetAttribute((const void*)k_build_A,
                      hipFuncAttributeMaxDynamicSharedMemorySize, (int)sm2);
  k_build_A<<<256, 256, sm2, stream>>>(Ur, Ui, A);

  size_t sm3 = (size_t)(128 * PSI_S + 16 * BT_S) * sizeof(float);  // 149,760 B
  hipFuncSetAttribute((const void*)k_qcnn_main,
                      hipFuncAttributeMaxDynamicSharedMemorySize, (int)sm3);
  k_qcnn_main<<<512, 256, sm3, stream>>>(x, A, weight, out);
}